// CrossAttention_robotcar_56968446214805
// MI455X (gfx1250) — compile-verified
//
#include <hip/hip_runtime.h>

#define BS   2
#define NQ   4704      // H1*BEV = 49*96
#define NV   768       // 24*32
#define E    128
#define NH   8
#define NP   4
#define HD   16
#define HFm  24
#define WFm  32
#define HRESm 16
#define PRE_PITCH 132  // mult-of-4 pitch: 16B-aligned stores, banks 4m+k (conflict-free)

typedef float v2f __attribute__((ext_vector_type(2)));
typedef float v4f __attribute__((ext_vector_type(4)));
typedef float v8f __attribute__((ext_vector_type(8)));
typedef unsigned int u32x4 __attribute__((ext_vector_type(4)));
typedef int i32x4 __attribute__((ext_vector_type(4)));
typedef int i32x8 __attribute__((ext_vector_type(8)));

__device__ __forceinline__ v8f wmma4(v2f a, v2f b, v8f c) {
  // D = A(16x4 f32) * B(4x16 f32) + C(16x16 f32), full fp32 precision
  return __builtin_amdgcn_wmma_f32_16x16x4_f32(false, a, false, b, (short)0, c,
                                               false, false);
}

// ---------------------------------------------------------------------------
// K1: v = value @ Wv + bv  -> vimg[(b*8+h)][y][x][d]   (1536 x 128 x 128 GEMM)
// one wave per 16x16 output tile
// ---------------------------------------------------------------------------
__global__ void k_vproj(const float* __restrict__ value, const float* __restrict__ Wv,
                        const float* __restrict__ bv, float* __restrict__ vimg) {
  const int mt = blockIdx.x;          // 0..95
  const int nt = blockIdx.y;          // 0..7
  const int lane = threadIdx.x;       // 0..31
  const int n  = lane & 15;
  const int kh = (lane >> 4) << 1;    // 0 or 2
  const int mrow = mt * 16 + n;       // A-fragment row (M = lane&15 in both halves)
  const int c = nt * 16 + n;          // output column
  v8f acc = {};
  const float* arow = value + mrow * E;
  for (int k0 = 0; k0 < E; k0 += 4) {
    v2f a, b;
    a = *(const v2f*)(arow + k0 + kh);
    b.x = Wv[(k0 + kh) * E + c];
    b.y = Wv[(k0 + kh + 1) * E + c];
    acc = wmma4(a, b, acc);
  }
  const float bias = bv[c];
  const int h = c >> 4, d = c & 15;
  const int bb = (mt >= 48) ? 1 : 0;          // block-uniform batch (48 tiles/batch)
  const int rbase = mt * 16 - bb * NV;        // row within batch, no per-lane div
  const int mhalf = (lane >> 4) << 3;
  for (int j = 0; j < 8; ++j) {
    const int idx = rbase + mhalf + j;        // row in nv = y*32+x
    const int y = idx >> 5, x = idx & 31;
    vimg[(((bb * NH + h) * HFm + y) * WFm + x) * HD + d] = acc[j] + bias;
  }
}

// ---------------------------------------------------------------------------
// K2: off = q@Woff+boff (N=64) and att-logits = q@Watt+batt (N=32), fused as
// 6 column tiles of a 9408 x 96 x 128 GEMM. Softmax over P deferred to K3.
// ---------------------------------------------------------------------------
__global__ void k_offatt(const float* __restrict__ query, const float* __restrict__ Woff,
                         const float* __restrict__ boff, const float* __restrict__ Watt,
                         const float* __restrict__ batt, float* __restrict__ off_ws,
                         float* __restrict__ attl_ws) {
  const int mt = blockIdx.x;          // 0..587
  const int nt = blockIdx.y;          // 0..5 (0-3: Woff, 4-5: Watt) -- uniform
  const int lane = threadIdx.x;
  const int n  = lane & 15;
  const int kh = (lane >> 4) << 1;
  const int mrow = mt * 16 + n;
  const float *B, *bias;
  float* dst;
  int ncols, cb;
  if (nt < 4) { B = Woff; bias = boff; dst = off_ws;  ncols = 64; cb = nt * 16; }
  else        { B = Watt; bias = batt; dst = attl_ws; ncols = 32; cb = (nt - 4) * 16; }
  const int c = cb + n;
  v8f acc = {};
  const float* arow = query + mrow * E;
  for (int k0 = 0; k0 < E; k0 += 4) {
    v2f a, b;
    a = *(const v2f*)(arow + k0 + kh);
    b.x = B[(k0 + kh) * ncols + c];
    b.y = B[(k0 + kh + 1) * ncols + c];
    acc = wmma4(a, b, acc);
  }
  const float bc = bias[c];
  const int mhalf = (lane >> 4) << 3;
  for (int j = 0; j < 8; ++j)
    dst[(mt * 16 + mhalf + j) * ncols + c] = acc[j] + bc;
}

// ---------------------------------------------------------------------------
// K3: fully fused msda(16 planes) + Wout GEMM + Wproj softmax/argmax/reduce.
// One 256-thread block per (batch, 16-query tile). grd3d never touches HBM:
// all 16 HRES planes (16q x 128e) live in 128KB of CDNA5's 320KB LDS.
// Wout staged via Tensor Data Mover (TENSORcnt), with a plain-copy hedge.
// ---------------------------------------------------------------------------
__global__ void k_main(const float* __restrict__ query, const float* __restrict__ u,
                       const float* __restrict__ v, const float* __restrict__ Wout,
                       const float* __restrict__ bout, const float* __restrict__ Wproj,
                       const float* __restrict__ bproj, const float* __restrict__ vimg,
                       const float* __restrict__ off_ws, const float* __restrict__ attl_ws,
                       float* __restrict__ out_bev, float* __restrict__ out_idx) {
  extern __shared__ float smem[];
  float* s_wout = smem;                       // 128*128      = 16384 f  (LDS offset 0)
  float* s_pre  = s_wout + E * E;             // 16*PRE_PITCH =  2112 f
  float* s_grd  = s_pre + 16 * PRE_PITCH;     // 16*16*128    = 32768 f
  float* s_wl   = s_grd + 16 * 16 * E;        // wl[q][rp]    =   256 f
  float* s_w    = s_wl + 256;                 // w[q][rp]     =   256 f
  float* s_wp   = s_w + 256;                  // Wproj        =   128 f

  const int t  = threadIdx.x;                 // 0..255 (8 waves)
  const int b  = blockIdx.y;
  const int qt = blockIdx.x;                  // 16-query tile, 0..293
  const float* qrow0 = query + (b * NQ + qt * 16) * E;

  __builtin_prefetch(qrow0 + t * 8, 0, 1);    // global_prefetch_b8 on query tile

  // ---- TDM: DMA Wout (64KB, 1-D tile of 16384 f32) into LDS base ----------
  if (t < 32) {                               // wave 0 issues once (EXEC ignored)
    const unsigned long long ga = (unsigned long long)(const void*)Wout;
    u32x4 g0;
    g0.x = 1u;                                        // count=1 valid descriptor
    g0.y = 0u;                                        // lds_addr = 0 (s_wout)
    g0.z = (unsigned)(ga & 0xffffffffu);              // global_addr[31:0]
    g0.w = (unsigned)((ga >> 32) & 0x01ffffffu)       // global_addr[56:32]
         | (2u << 30);                                // type = 2 ("image")
    i32x8 g1;
    g1[0] = (int)(2u << 16);                          // data_size = 4 bytes
    g1[1] = (int)(0x4000u << 16);                     // tensor_dim0 = 16384 (lo16)
    g1[2] = (int)(1u << 16);                          // dim0 hi=0 | tensor_dim1 = 1
    g1[3] = (int)(0x4000u << 16);                     // tile_dim0 = 16384
    g1[4] = 0;                                        // tile_dim1 = tile_dim2 = 0
    g1[5] = 0x4000;                                   // tensor_dim0_stride = 16384
    g1[6] = 0; g1[7] = 0;
    i32x4 z4 = {0, 0, 0, 0};                          // groups 2/3 unused (<=2D)
    i32x8 z8 = {0, 0, 0, 0, 0, 0, 0, 0};              // extra group (clang-23 form)
    __builtin_amdgcn_tensor_load_to_lds(g0, g1, z4, z4, z8, 0);
    __builtin_amdgcn_s_wait_tensorcnt(0);
  }
  __syncthreads();
  // hedge: builtin operand semantics are arity-verified only; plain copy is
  // ~64KB from warm L2 per WG (negligible) and guarantees s_wout correctness.
  for (int i = t; i < E * E; i += 256) s_wout[i] = Wout[i];
  if (t < E) s_wp[t] = Wproj[t];
  __syncthreads();

  // ---- sampling mapping: t -> (query, head, half of HD) ----
  const int sq = t >> 4;                      // 0..15
  const int sh = (t >> 1) & 7;                // 0..7
  const int d0 = (t & 1) << 3;                // 0 or 8
  const int qg = qt * 16 + sq;
  const int row = b * NQ + qg;

  // attention softmax over P (rp-invariant)
  float aw[NP];
  {
    const float* al = attl_ws + row * (NH * NP) + sh * NP;
    float m = al[0];
    for (int p = 1; p < NP; ++p) m = fmaxf(m, al[p]);
    float s = 0.f;
    for (int p = 0; p < NP; ++p) { aw[p] = __expf(al[p] - m); s += aw[p]; }
    const float inv = 1.f / s;
    for (int p = 0; p < NP; ++p) aw[p] *= inv;
  }
  float offc[NP][2];
  {
    const float* op = off_ws + row * (NH * NP * 2) + sh * (NP * 2);
    for (int p = 0; p < NP; ++p) { offc[p][0] = op[2 * p]; offc[p][1] = op[2 * p + 1]; }
  }
  const float* img = vimg + (b * NH + sh) * (HFm * WFm * HD);

  // ---- GEMM mapping: wave -> 16-col tile of Wout ----
  const int lane = t & 31;
  const int wv = t >> 5;                      // 0..7
  const int gn  = lane & 15;
  const int gkh = (lane >> 4) << 1;
  const int gc  = wv * 16 + gn;
  const int gmh = (lane >> 4) << 3;
  const float bo = bout[gc];

  for (int rp = 0; rp < HRESm; ++rp) {
    // ---- bilinear deformable sampling into s_pre (16q x 128e) ----
    v4f acc0 = {}, acc1 = {};
    const float ru = u[qg * HRESm + rp];
    const float rv = v[qg * HRESm + rp];
    auto tap = [&](int xi, int yi, float wt) {
      if ((unsigned)xi < (unsigned)WFm && (unsigned)yi < (unsigned)HFm) {
        const v4f* pv = (const v4f*)(img + (yi * WFm + xi) * HD + d0); // 32B aligned
        acc0 += wt * pv[0];
        acc1 += wt * pv[1];
      }
    };
    for (int p = 0; p < NP; ++p) {
      const float x = (ru + offc[p][0] * (1.0f / WFm)) * WFm - 0.5f;
      const float y = (rv + offc[p][1] * (1.0f / HFm)) * HFm - 0.5f;
      const float x0 = floorf(x), y0 = floorf(y);
      const float wx = x - x0, wy = y - y0;
      const int xi = (int)x0, yi = (int)y0;
      tap(xi,     yi,     (1.f - wx) * (1.f - wy) * aw[p]);
      tap(xi + 1, yi,     wx * (1.f - wy) * aw[p]);
      tap(xi,     yi + 1, (1.f - wx) * wy * aw[p]);
      tap(xi + 1, yi + 1, wx * wy * aw[p]);
    }
    {
      v4f* dst = (v4f*)(s_pre + sq * PRE_PITCH + sh * HD + d0);        // 16B aligned
      dst[0] = acc0;
      dst[1] = acc1;
    }
    __syncthreads();

    // ---- grd[rp] = s_pre @ Wout + bout + query  (8 waves, WMMA f32) ----
    v8f gacc = {};
    for (int k0 = 0; k0 < E; k0 += 4) {
      v2f a = *(const v2f*)(s_pre + gn * PRE_PITCH + k0 + gkh);        // ds_load_b64
      v2f bfr;
      bfr.x = s_wout[(k0 + gkh) * E + gc];
      bfr.y = s_wout[(k0 + gkh + 1) * E + gc];
      gacc = wmma4(a, bfr, gacc);
    }
    for (int j = 0; j < 8; ++j)
      s_grd[(rp * 16 + gmh + j) * E + gc] = gacc[j] + bo + qrow0[(gmh + j) * E + gc];
    __syncthreads();
  }

  // ---- w-logits: wl[q][rp] = grd . Wproj + bproj ----
  {
    const int qq = t & 15, rp = t >> 4;
    const v4f* g4 = (const v4f*)(s_grd + (rp * 16 + qq) * E);
    const v4f* w4 = (const v4f*)s_wp;
    v4f sv = {};
    for (int i = 0; i < E / 4; ++i) sv += g4[i] * w4[i];
    s_wl[qq * 16 + rp] = sv.x + sv.y + sv.z + sv.w + bproj[0];
  }
  __syncthreads();

  // ---- softmax over HRES + argmax (first-max, matches jnp.argmax) ----
  if (t < 16) {
    const float* wl = s_wl + t * 16;
    float m = wl[0];
    int am = 0;
    for (int rp = 1; rp < HRESm; ++rp)
      if (wl[rp] > m) { m = wl[rp]; am = rp; }
    float s = 0.f;
    for (int rp = 0; rp < HRESm; ++rp) { float e = __expf(wl[rp] - m); s_w[t * 16 + rp] = e; s += e; }
    const float inv = 1.f / s;
    for (int rp = 0; rp < HRESm; ++rp) s_w[t * 16 + rp] *= inv;
    out_idx[b * NQ + qt * 16 + t] = (float)am;
  }
  __syncthreads();

  // ---- bev = sum_rp w*grd + query ----
  {
    const int qq = t >> 4;
    const int e0 = (t & 15) * 8;
    const int rowq = b * NQ + qt * 16 + qq;
    v4f s0 = {}, s1 = {};
    for (int rp = 0; rp < HRESm; ++rp) {
      const float w = s_w[qq * 16 + rp];
      const v4f* g4 = (const v4f*)(s_grd + (rp * 16 + qq) * E + e0);
      s0 += w * g4[0];
      s1 += w * g4[1];
    }
    const v4f* q4 = (const v4f*)(query + rowq * E + e0);
    v4f* o4 = (v4f*)(out_bev + rowq * E + e0);
    o4[0] = s0 + q4[0];
    o4[1] = s1 + q4[1];
  }
}

// ---------------------------------------------------------------------------
extern "C" void kernel_launch(void* const* d_in, const int* in_sizes, int n_in,
                              void* d_out, int out_size, void* d_ws, size_t ws_size,
                              hipStream_t stream) {
  (void)in_sizes; (void)n_in; (void)out_size; (void)ws_size;
  const float* query = (const float*)d_in[0];
  const float* value = (const float*)d_in[1];
  const float* u     = (const float*)d_in[2];
  const float* v     = (const float*)d_in[3];
  const float* Wv    = (const float*)d_in[4];
  const float* bv    = (const float*)d_in[5];
  const float* Woff  = (const float*)d_in[6];
  const float* boff  = (const float*)d_in[7];
  const float* Watt  = (const float*)d_in[8];
  const float* batt  = (const float*)d_in[9];
  const float* Wout  = (const float*)d_in[10];
  const float* bout  = (const float*)d_in[11];
  const float* Wproj = (const float*)d_in[12];
  const float* bproj = (const float*)d_in[13];

  float* ws      = (float*)d_ws;
  float* vimg    = ws;                                       // 196608 f
  float* off_ws  = vimg + (size_t)BS * NH * HFm * WFm * HD;  // 602112 f
  float* attl_ws = off_ws + (size_t)BS * NQ * NH * NP * 2;   // 301056 f

  float* out_bev = (float*)d_out;
  float* out_idx = out_bev + (size_t)BS * NQ * E;            // argmax as float

  k_vproj <<<dim3(96, 8), 32, 0, stream>>>(value, Wv, bv, vimg);
  k_offatt<<<dim3(BS * NQ / 16, 6), 32, 0, stream>>>(query, Woff, boff, Watt, batt,
                                                     off_ws, attl_ws);
  const size_t lds = (size_t)(E * E + 16 * PRE_PITCH + 16 * 16 * E + 256 + 256 + E)
                     * sizeof(float);
  k_main  <<<dim3(NQ / 16, BS), 256, lds, stream>>>(query, u, v, Wout, bout, Wproj, bproj,
                                                    vimg, off_ws, attl_ws, out_bev, out_idx);
}